// BatchedFerroelectricBasis_53283364274897
// MI455X (gfx1250) — compile-verified
//
#include <hip/hip_runtime.h>

typedef __attribute__((ext_vector_type(2))) float v2f;
typedef __attribute__((ext_vector_type(8))) float v8f;

#define GATE_SLOPE 10.0f
#define ALPHA      0.8f
#define B_DIM      256
#define IN_DIM     128
#define OUT_DIM    128
#define NUM_BASIS  16
#define ELEMS      8     // state elements per thread (n-values)
#define BT         16    // batch tile (rows per WMMA reduction)

__device__ __forceinline__ float fast_rcp(float x) {
#if __has_builtin(__builtin_amdgcn_rcpf)
    return __builtin_amdgcn_rcpf(x);
#else
    return 1.0f / x;
#endif
}

__device__ __forceinline__ float fast_sigmoid(float z) {
    // 1 / (1 + e^-z); overflow of exp -> rcp(inf) = 0, correct saturation
    return fast_rcp(1.0f + __expf(-z));
}

__device__ __forceinline__ float fast_tanh(float z) {
#if __has_builtin(__builtin_amdgcn_tanhf)
    return __builtin_amdgcn_tanhf(z);   // gfx1250 v_tanh_f32
#else
    float e = __expf(-2.0f * z);
    return (1.0f - e) * fast_rcp(1.0f + e);
#endif
}

__global__ __launch_bounds__(256) void ferro_basis_kernel(
    const float* __restrict__ x,    // (B, IN_DIM)
    const float* __restrict__ k,    // (IN_DIM, OUT_DIM, NUM_BASIS)
    const float* __restrict__ Ec,
    const float* __restrict__ Ps,
    const float* __restrict__ bias,
    const float* __restrict__ coef,
    float* __restrict__ out)        // (B, OUT_DIM)
{
    const int o    = blockIdx.x;          // one output column per workgroup
    const int t    = threadIdx.x;
    const int i    = t & (IN_DIM - 1);    // input row owned by this thread
    const int ngrp = t >> 7;              // 0 or 1 -> which half of the 16 n's
    const int lane = t & 31;
    const int wave = t >> 5;              // 8 wave32s per block

    __shared__ float xtile[BT * IN_DIM];  // 16 rows of x (8 KB)
    __shared__ float pred[BT * 9];        // 16 x 8 wave partials, stride-9 padded

    // ---- load per-thread parameters (8 consecutive n values, 32B aligned) ----
    const int pbase = (i * OUT_DIM + o) * NUM_BASIS + ngrp * ELEMS;
    float pk[ELEMS], pEc[ELEMS], pPs[ELEMS], pB[ELEMS], pC[ELEMS], bs[ELEMS];
#pragma unroll
    for (int e = 0; e < ELEMS; e += 4) {
        float4 a = *(const float4*)(k    + pbase + e);
        float4 b = *(const float4*)(Ec   + pbase + e);
        float4 c = *(const float4*)(Ps   + pbase + e);
        float4 d = *(const float4*)(bias + pbase + e);
        float4 f = *(const float4*)(coef + pbase + e);
        pk[e+0]=a.x; pk[e+1]=a.y; pk[e+2]=a.z; pk[e+3]=a.w;
        pEc[e+0]=b.x; pEc[e+1]=b.y; pEc[e+2]=b.z; pEc[e+3]=b.w;
        pPs[e+0]=c.x; pPs[e+1]=c.y; pPs[e+2]=c.z; pPs[e+3]=c.w;
        pB[e+0]=d.x; pB[e+1]=d.y; pB[e+2]=d.z; pB[e+3]=d.w;
        pC[e+0]=f.x; pC[e+1]=f.y; pC[e+2]=f.z; pC[e+3]=f.w;
    }
#pragma unroll
    for (int e = 0; e < ELEMS; ++e) bs[e] = 1.0f;   // init carry = ones

    float xprev = 0.0f;                              // init prev_x = zeros
    const v2f bones = {1.0f, 1.0f};                  // ones B-matrix (all elements 1)

    for (int tile = 0; tile < B_DIM / BT; ++tile) {
        const int b0 = tile * BT;

        // ---- stage 16 rows of x into LDS (coalesced float4) ----
        {
            const float4* src = (const float4*)(x + b0 * IN_DIM);
            float4* dst = (float4*)xtile;
            dst[t * 2 + 0] = src[t * 2 + 0];
            dst[t * 2 + 1] = src[t * 2 + 1];
        }
        __syncthreads();

#pragma unroll 1
        for (int bt = 0; bt < BT; ++bt) {
            const float xs = xtile[bt * IN_DIM + i];
            // is_up depends only on (b, i): one exp shared by all 8 elements
            const float is_up = fast_sigmoid(GATE_SLOPE * (xs - xprev));
            const float om_up = 1.0f - is_up;
            float acc = 0.0f;
#pragma unroll
            for (int e = 0; e < ELEMS; ++e) {
                const float ec  = pEc[e];
                const float cp  = fast_sigmoid(GATE_SLOPE * ( xs - ec));
                const float cn  = fast_sigmoid(GATE_SLOPE * (-xs - ec));
                const float su  = is_up * cp;
                const float sl  = om_up * cn;
                const float tgt = su - sl + (1.0f - su - sl) * bs[e];
                const float bn  = ALPHA * bs[e] + (1.0f - ALPHA) * tgt;
                bs[e] = bn;
                const float basis = pPs[e] * fast_tanh(pk[e] * (xs + ec * bn)) + pB[e];
                acc = fmaf(basis, pC[e], acc);
            }
            xprev = xs;

            // wave32 tree reduce -> one partial per wave
#pragma unroll
            for (int m = 16; m >= 1; m >>= 1) acc += __shfl_xor(acc, m, 32);
            if (lane == 0) pred[bt * 9 + wave] = acc;
        }
        __syncthreads();

        // ---- WMMA reduction: row-sums of pred[16][8] via D = A x ones + C ----
        if (wave == 0) {
            // A-matrix f32 16x4 layout: lanes 0-15 -> (M=lane, K=v+0),
            //                           lanes16-31 -> (M=lane-16, K=v+2)
            const int row = lane & 15;
            const int kk  = (lane >> 4) * 2;
            v2f a0, a1;
            a0.x = pred[row * 9 + kk + 0];
            a0.y = pred[row * 9 + kk + 1];
            a1.x = pred[row * 9 + 4 + kk + 0];
            a1.y = pred[row * 9 + 4 + kk + 1];
            v8f d = {};
            d = __builtin_amdgcn_wmma_f32_16x16x4_f32(false, a0, false, bones,
                                                      (short)0, d, false, false);
            d = __builtin_amdgcn_wmma_f32_16x16x4_f32(false, a1, false, bones,
                                                      (short)0, d, false, false);
            // C/D layout: VGPR r: lanes0-15 -> M=r (N=lane), lanes16-31 -> M=8+r
            if (lane == 0) {
#pragma unroll
                for (int r = 0; r < 8; ++r) out[(b0 + r) * OUT_DIM + o] = d[r];
            } else if (lane == 16) {
#pragma unroll
                for (int r = 0; r < 8; ++r) out[(b0 + 8 + r) * OUT_DIM + o] = d[r];
            }
        }
        __syncthreads();   // protects pred/xtile reuse next tile
    }
}

extern "C" void kernel_launch(void* const* d_in, const int* in_sizes, int n_in,
                              void* d_out, int out_size, void* d_ws, size_t ws_size,
                              hipStream_t stream) {
    (void)in_sizes; (void)n_in; (void)out_size; (void)d_ws; (void)ws_size;
    const float* x    = (const float*)d_in[0];
    const float* k    = (const float*)d_in[1];
    const float* Ec   = (const float*)d_in[2];
    const float* Ps   = (const float*)d_in[3];
    const float* bias = (const float*)d_in[4];
    const float* coef = (const float*)d_in[5];
    float* out        = (float*)d_out;

    ferro_basis_kernel<<<OUT_DIM, 256, 0, stream>>>(x, k, Ec, Ps, bias, coef, out);
}